// FNetBasicFourierTransform_55327768708130
// MI455X (gfx1250) — compile-verified
//
#include <hip/hip_runtime.h>
#include <math.h>

// FNet 2D-FFT real part as two DFT GEMM passes on f32 WMMA (gfx1250), with
// double-buffered async global->LDS staging (GLOBAL_LOAD_ASYNC_TO_LDS_*).
//   Init   : materialize C_h, S_h (768x768) in workspace (L2-resident)
//   Stage A: Tc = X @ C_h, Ts = X @ S_h              (K = 768)
//   Stage B: Out = C_s @ Tc - S_s @ Ts               (K = 4096, -S baked into LDS table)
// B tiles land TRANSPOSED in LDS (async B32 scatter) so each WMMA B-fragment
// is one aligned ds_load_b64. Each wave computes 32x64 (2 M x 4 N subtiles).

typedef __attribute__((ext_vector_type(2))) float f32x2;
typedef __attribute__((ext_vector_type(8))) float f32x8;

#define SEQ   4096
#define HID   768
#define BATCH 8
#define KC    32            // K chunk per LDS stage
#define KP    (KC + 4)      // padded K stride (floats) for tiles
#define TWO_PI 6.28318530717958647692f

static __device__ __forceinline__ f32x8 wmma_f32(f32x2 a, f32x2 b, f32x8 c) {
    return __builtin_amdgcn_wmma_f32_16x16x4_f32(false, a, false, b, (short)0, c, false, false);
}

// --- CDNA5 async global->LDS copies (ASYNCcnt-tracked DMA, no VGPR round trip)
static __device__ __forceinline__ void async_b32(unsigned lds, unsigned long long g) {
    asm volatile("global_load_async_to_lds_b32 %0, %1, off" :: "v"(lds), "v"(g) : "memory");
}
static __device__ __forceinline__ void async_b128(unsigned lds, unsigned long long g) {
    asm volatile("global_load_async_to_lds_b128 %0, %1, off" :: "v"(lds), "v"(g) : "memory");
}
static __device__ __forceinline__ void wait_async0() {
    asm volatile("s_wait_asynccnt 0x0" ::: "memory");
}
// Generic->LDS byte offset: LDS aperture uses addr[31:0] as the LDS address.
static __device__ __forceinline__ unsigned lds_off(const void* p) {
    return (unsigned)(unsigned long long)p;
}

// ---------------------------------------------------------------------------
// Init: C_h[k][j] = cos(2*pi*k*j/768), S_h[k][j] = sin(...)
// ---------------------------------------------------------------------------
__global__ __launch_bounds__(256) void fnet_make_hidden_dft(
    float* __restrict__ Ch, float* __restrict__ Sh)
{
    const unsigned i = blockIdx.x * 256 + threadIdx.x;   // < 768*768
    const unsigned k = i / HID;
    const unsigned j = i - k * HID;
    const unsigned m = (k * j) % HID;
    float s, c;
    __sincosf((TWO_PI / (float)HID) * (float)m, &s, &c);
    Ch[i] = c;
    Sh[i] = s;
}

// ---------------------------------------------------------------------------
// Stage A: rows = BATCH*SEQ = 32768 flattened. Block tile 256(M) x 64(N),
// 8 waves, each wave 32x64. A (=X) row-major; B (=C_h/S_h) transposed.
// ---------------------------------------------------------------------------
__global__ __launch_bounds__(256) void fnet_hidden_dft(
    const float* __restrict__ X,
    const float* __restrict__ Ch, const float* __restrict__ Sh,
    float* __restrict__ Tc, float* __restrict__ Ts)
{
    __shared__ float Atile[2][256][KP];   // 2 x 36.9 KB
    __shared__ float Bct[2][64][KP];      // transposed [col][k], 2 x 9.2 KB
    __shared__ float Bst[2][64][KP];

    const int tid  = threadIdx.x;
    const int lane = tid & 31;
    const int wave = tid >> 5;
    const int l16  = lane & 15;
    const int half = lane >> 4;           // 0: K{0,1}, 1: K{2,3}

    const int    nBase   = blockIdx.x * 64;
    const size_t rowBase = (size_t)blockIdx.y * 256;

    // Per-thread staging coordinates
    const int ar  = tid >> 3;             // A row 0..31 (+32 per step)
    const int akq = (tid & 7) * 4;        // A k quad
    const int bc_ = tid & 63;             // B column
    const int br  = tid >> 6;             // B k row 0..3 (+4 per step)

    f32x8 accC[2][4], accS[2][4];
    #pragma unroll
    for (int m = 0; m < 2; ++m)
        #pragma unroll
        for (int t = 0; t < 4; ++t) { accC[m][t] = (f32x8){}; accS[m][t] = (f32x8){}; }

    // Issue async staging for chunk k0 into buffer `buf`
    auto issue = [&](int buf, int k0) {
        unsigned la = lds_off(&Atile[buf][ar][akq]);
        unsigned long long ga = (unsigned long long)(X + (rowBase + ar) * HID + k0 + akq);
        #pragma unroll
        for (int v = 0; v < 8; ++v) {                       // 256x32 f32, b128 each
            async_b128(la, ga);
            la += 32u * KP * 4;  ga += 32ull * HID * 4;     // +32 rows
        }
        unsigned lbc = lds_off(&Bct[buf][bc_][br]);
        unsigned lbs = lds_off(&Bst[buf][bc_][br]);
        unsigned long long gc = (unsigned long long)(Ch + (size_t)(k0 + br) * HID + nBase + bc_);
        unsigned long long gs = (unsigned long long)(Sh + (size_t)(k0 + br) * HID + nBase + bc_);
        #pragma unroll
        for (int v = 0; v < 8; ++v) {                       // 32x64 f32 each, b32 scatter-transpose
            async_b32(lbc, gc);
            async_b32(lbs, gs);
            lbc += 16; lbs += 16;                           // +4 k
            gc += 4ull * HID * 4; gs += 4ull * HID * 4;
        }
    };

    issue(0, 0);
    for (int k0 = 0; k0 < HID; k0 += KC) {
        const int buf = (k0 / KC) & 1;
        wait_async0();
        __syncthreads();                  // all copies landed; all waves done with buf^1
        if (k0 + KC < HID) issue(buf ^ 1, k0 + KC);

        #pragma unroll
        for (int kk = 0; kk < KC; kk += 4) {
            const int ks = kk + half * 2;
            f32x2 a0, a1;
            a0.x = Atile[buf][wave * 32 + l16][ks];      a0.y = Atile[buf][wave * 32 + l16][ks + 1];
            a1.x = Atile[buf][wave * 32 + 16 + l16][ks]; a1.y = Atile[buf][wave * 32 + 16 + l16][ks + 1];

            #pragma unroll
            for (int t = 0; t < 4; ++t) {
                const int col = t * 16 + l16;
                f32x2 bc, bs;
                bc.x = Bct[buf][col][ks]; bc.y = Bct[buf][col][ks + 1];   // one ds_load_b64
                bs.x = Bst[buf][col][ks]; bs.y = Bst[buf][col][ks + 1];
                accC[0][t] = wmma_f32(a0, bc, accC[0][t]);
                accC[1][t] = wmma_f32(a1, bc, accC[1][t]);
                accS[0][t] = wmma_f32(a0, bs, accS[0][t]);
                accS[1][t] = wmma_f32(a1, bs, accS[1][t]);
            }
        }
    }

    // D layout: VGPR r -> M = r + 8*half; N = l16
    #pragma unroll
    for (int m = 0; m < 2; ++m) {
        const size_t row0 = rowBase + wave * 32 + m * 16 + 8 * half;
        #pragma unroll
        for (int t = 0; t < 4; ++t) {
            const int col = nBase + t * 16 + l16;
            #pragma unroll
            for (int r = 0; r < 8; ++r) {
                Tc[(row0 + r) * HID + col] = accC[m][t][r];
                Ts[(row0 + r) * HID + col] = accS[m][t][r];
            }
        }
    }
}

// ---------------------------------------------------------------------------
// Stage B: per batch, Out = C_s @ Tc + (-S_s) @ Ts. Block tile 256(M) x 64(N),
// K = 4096. Tc/Ts tiles async-staged transposed; A fragments from 4096-entry
// cos / (-sin) LDS tables (mul + and).
// ---------------------------------------------------------------------------
__global__ __launch_bounds__(256) void fnet_seq_dft(
    const float* __restrict__ Tc, const float* __restrict__ Ts,
    float* __restrict__ Out)
{
    __shared__ float cosT[SEQ];           // 16 KB
    __shared__ float msinT[SEQ];          // 16 KB (negated sin)
    __shared__ float Bct[2][64][KP];      // transposed Tc tile, double-buffered
    __shared__ float Bst[2][64][KP];

    const int tid  = threadIdx.x;
    const int lane = tid & 31;
    const int wave = tid >> 5;
    const int l16  = lane & 15;
    const int half = lane >> 4;

    for (int i = tid; i < SEQ; i += 256) {
        float s, c;
        __sincosf((TWO_PI / (float)SEQ) * (float)i, &s, &c);
        cosT[i]  = c;
        msinT[i] = -s;
    }

    const int    b     = blockIdx.z;
    const int    mBase = blockIdx.y * 256;
    const int    nBase = blockIdx.x * 64;
    const float* tc    = Tc + (size_t)b * SEQ * HID;
    const float* ts    = Ts + (size_t)b * SEQ * HID;

    const int bc_ = tid & 63;             // B column
    const int br  = tid >> 6;             // B k row 0..3 (+4 per step)

    auto issue = [&](int buf, int k0) {
        unsigned lbc = lds_off(&Bct[buf][bc_][br]);
        unsigned lbs = lds_off(&Bst[buf][bc_][br]);
        unsigned long long gc = (unsigned long long)(tc + (size_t)(k0 + br) * HID + nBase + bc_);
        unsigned long long gs = (unsigned long long)(ts + (size_t)(k0 + br) * HID + nBase + bc_);
        #pragma unroll
        for (int v = 0; v < 8; ++v) {
            async_b32(lbc, gc);
            async_b32(lbs, gs);
            lbc += 16; lbs += 16;
            gc += 4ull * HID * 4; gs += 4ull * HID * 4;
        }
    };

    f32x8 acc[2][4];
    #pragma unroll
    for (int m = 0; m < 2; ++m)
        #pragma unroll
        for (int t = 0; t < 4; ++t) acc[m][t] = (f32x8){};

    const unsigned arow0 = (unsigned)(mBase + wave * 32 + l16);
    const unsigned arow1 = arow0 + 16;

    issue(0, 0);
    for (int k0 = 0; k0 < SEQ; k0 += KC) {
        const int buf = (k0 / KC) & 1;
        wait_async0();
        __syncthreads();                  // copies landed; tables visible; buf^1 free
        if (k0 + KC < SEQ) issue(buf ^ 1, k0 + KC);

        #pragma unroll
        for (int kk = 0; kk < KC; kk += 4) {
            const int      ks = kk + half * 2;
            const unsigned k  = (unsigned)(k0 + ks);
            unsigned p0 = (arow0 * k) & (SEQ - 1);
            unsigned p1 = (p0 + arow0) & (SEQ - 1);   // arow0*(k+1) mod SEQ
            unsigned q0 = (arow1 * k) & (SEQ - 1);
            unsigned q1 = (q0 + arow1) & (SEQ - 1);
            f32x2 ac0, as0, ac1, as1;
            ac0.x = cosT[p0];  ac0.y = cosT[p1];
            as0.x = msinT[p0]; as0.y = msinT[p1];
            ac1.x = cosT[q0];  ac1.y = cosT[q1];
            as1.x = msinT[q0]; as1.y = msinT[q1];

            #pragma unroll
            for (int t = 0; t < 4; ++t) {
                const int col = t * 16 + l16;
                f32x2 bc, bs;
                bc.x = Bct[buf][col][ks]; bc.y = Bct[buf][col][ks + 1];   // one ds_load_b64
                bs.x = Bst[buf][col][ks]; bs.y = Bst[buf][col][ks + 1];
                acc[0][t] = wmma_f32(ac0, bc, acc[0][t]);
                acc[1][t] = wmma_f32(ac1, bc, acc[1][t]);
                acc[0][t] = wmma_f32(as0, bs, acc[0][t]);
                acc[1][t] = wmma_f32(as1, bs, acc[1][t]);
            }
        }
    }

    float* o = Out + (size_t)b * SEQ * HID;
    #pragma unroll
    for (int m = 0; m < 2; ++m) {
        const size_t row0 = (size_t)mBase + wave * 32 + m * 16 + 8 * half;
        #pragma unroll
        for (int t = 0; t < 4; ++t) {
            const int col = nBase + t * 16 + l16;
            #pragma unroll
            for (int r = 0; r < 8; ++r) {
                o[(row0 + r) * HID + col] = acc[m][t][r];
            }
        }
    }
}

// ---------------------------------------------------------------------------
extern "C" void kernel_launch(void* const* d_in, const int* in_sizes, int n_in,
                              void* d_out, int out_size, void* d_ws, size_t ws_size,
                              hipStream_t stream) {
    const float* X   = (const float*)d_in[0];
    float*       out = (float*)d_out;

    // Workspace layout: Ch | Sh | Tc | Ts  (~206 MB total)
    float* Ch = (float*)d_ws;
    float* Sh = Ch + (size_t)HID * HID;
    float* Tc = Sh + (size_t)HID * HID;
    float* Ts = Tc + (size_t)BATCH * SEQ * HID;

    fnet_make_hidden_dft<<<(HID * HID) / 256, 256, 0, stream>>>(Ch, Sh);

    dim3 gridA(HID / 64, (BATCH * SEQ) / 256, 1);   // (12, 128)
    fnet_hidden_dft<<<gridA, 256, 0, stream>>>(X, Ch, Sh, Tc, Ts);

    dim3 gridB(HID / 64, SEQ / 256, BATCH);         // (12, 16, 8)
    fnet_seq_dft<<<gridB, 256, 0, stream>>>(Tc, Ts, out);
}